// XLMRForTokenClassificationWithCRF_71322226917764
// MI455X (gfx1250) — compile-verified
//
#include <hip/hip_runtime.h>
#include <hip/hip_bf16.h>

typedef __attribute__((ext_vector_type(2))) float v2f;
typedef __attribute__((ext_vector_type(8))) float v8f;

#define Bn 128
#define Tn 512
#define Hn 1024
#define Ln 48

// ---------------------------------------------------------------------------
// Kernel 1: logits[b,t,l] = sum_h features[b,t,h]*head_w[l,h] + head_b[l]
// One wave -> one 16-row x 48-col tile, K=1024 via V_WMMA_F32_16X16X4_F32.
// 8 waves (256 threads) per block -> 128 rows per block, 512 blocks.
// ---------------------------------------------------------------------------
__global__ __launch_bounds__(256) void crf_logits_wmma(
    const float* __restrict__ features, const float* __restrict__ head_w,
    const float* __restrict__ head_b, float* __restrict__ logits)
{
    const int lane = threadIdx.x & 31;
    const int wave = threadIdx.x >> 5;
    const int n = lane & 15;   // row (A) / col (B) within 16-wide tile
    const int g = lane >> 4;   // K sub-group: K = elt + 2*g
    const long row0 = ((long)blockIdx.x * 8 + wave) * 16;

    const float* arow = features + (row0 + n) * Hn + 2 * g;
    const float* b0   = head_w + (long)(n     ) * Hn + 2 * g;
    const float* b1   = head_w + (long)(n + 16) * Hn + 2 * g;
    const float* b2   = head_w + (long)(n + 32) * Hn + 2 * g;

    v8f acc0 = {}; v8f acc1 = {}; v8f acc2 = {};

    for (int k0 = 0; k0 < Hn; k0 += 64) {
        // prefetch the next chunk of the 268MB feature stream
        __builtin_prefetch(arow + k0 + 64, 0, 0);
#pragma unroll
        for (int kk = 0; kk < 64; kk += 4) {
            const int k = k0 + kk;
            v2f a  = *(const v2f*)(arow + k);
            v2f w0 = *(const v2f*)(b0 + k);
            v2f w1 = *(const v2f*)(b1 + k);
            v2f w2 = *(const v2f*)(b2 + k);
            acc0 = __builtin_amdgcn_wmma_f32_16x16x4_f32(
                false, a, false, w0, (short)0, acc0, false, false);
            acc1 = __builtin_amdgcn_wmma_f32_16x16x4_f32(
                false, a, false, w1, (short)0, acc1, false, false);
            acc2 = __builtin_amdgcn_wmma_f32_16x16x4_f32(
                false, a, false, w2, (short)0, acc2, false, false);
        }
    }

    const float bias0 = head_b[n];
    const float bias1 = head_b[n + 16];
    const float bias2 = head_b[n + 32];
#pragma unroll
    for (int r = 0; r < 8; ++r) {
        // D layout: VGPR r, lane group g -> output row r + 8*g
        float* o = logits + (row0 + r + 8 * g) * Ln;
        o[n]      = acc0[r] + bias0;
        o[n + 16] = acc1[r] + bias1;
        o[n + 32] = acc2[r] + bias2;
    }
}

// ---------------------------------------------------------------------------
// Kernel 2: per-batch CRF forward algorithm + gold-path numerator.
// One block per batch; 192 threads = 4 groups x 48 states (no idle lanes).
// Each group-q reduces 12 of the 48 source states; partial (max,sum) pairs
// are combined in LDS -> 4x shorter dependent chain on the sequential scan.
// ---------------------------------------------------------------------------
__global__ __launch_bounds__(192) void crf_forward(
    const float* __restrict__ logits, const int* __restrict__ labels,
    const float* __restrict__ start_t, const float* __restrict__ end_t,
    const float* __restrict__ trans_g, float* __restrict__ loss_b)
{
    __shared__ float trans[Ln * Ln];   // 9216 B
    __shared__ float alpha[Ln];
    __shared__ float partm[192];
    __shared__ float parts[192];
    __shared__ float red[192];

    const int tid = threadIdx.x;
    const int b   = blockIdx.x;
    const int q   = tid / Ln;          // 0..3 : which 12-wide slice of i
    const int j   = tid - q * Ln;      // 0..47: target state
    const int i0  = q * 12;

    const float* lg  = logits + (long)b * Tn * Ln;
    const int*   lab = labels + b * Tn;

    for (int i = tid; i < Ln * Ln; i += 192) trans[i] = trans_g[i];
    if (tid < Ln) alpha[tid] = start_t[tid] + lg[tid];
    __syncthreads();

    for (int t = 1; t < Tn; ++t) {
        float m = -3.4e38f;
#pragma unroll
        for (int i = i0; i < i0 + 12; ++i)
            m = fmaxf(m, alpha[i] + trans[i * Ln + j]);
        float s = 0.f;
#pragma unroll
        for (int i = i0; i < i0 + 12; ++i)
            s += __expf(alpha[i] + trans[i * Ln + j] - m);
        partm[tid] = m;
        parts[tid] = s;
        __syncthreads();                 // all alpha reads done
        if (q == 0) {
            float m0 = partm[j],            m1 = partm[j + 48];
            float m2 = partm[j + 96],       m3 = partm[j + 144];
            float ma = fmaxf(fmaxf(m0, m1), fmaxf(m2, m3));
            float sa = parts[j]       * __expf(m0 - ma)
                     + parts[j + 48]  * __expf(m1 - ma)
                     + parts[j + 96]  * __expf(m2 - ma)
                     + parts[j + 144] * __expf(m3 - ma);
            alpha[j] = ma + __logf(sa) + lg[t * Ln + j];
        }
        __syncthreads();                 // new alpha visible
    }

    // ---- numerator partials: gold emissions + gold transitions ----
    float acc = 0.f;
    for (int t = tid; t < Tn; t += 192)     acc += lg[t * Ln + lab[t]];
    for (int t = tid; t < Tn - 1; t += 192) acc += trans[lab[t] * Ln + lab[t + 1]];
    red[tid] = acc;
    __syncthreads();

    if (tid == 0) {
        float m = -3.4e38f;
        for (int jj = 0; jj < Ln; ++jj) m = fmaxf(m, alpha[jj] + end_t[jj]);
        float s = 0.f;
        for (int jj = 0; jj < Ln; ++jj) s += __expf(alpha[jj] + end_t[jj] - m);
        const float logZ = m + __logf(s);
        float num = start_t[lab[0]] + end_t[lab[Tn - 1]];
        for (int i = 0; i < 192; ++i) num += red[i];
        loss_b[b] = logZ - num;
    }
}

// ---------------------------------------------------------------------------
// Kernel 3: mean over the 128 per-batch losses (deterministic, no atomics).
// ---------------------------------------------------------------------------
__global__ __launch_bounds__(128) void crf_reduce(
    const float* __restrict__ loss_b, float* __restrict__ out)
{
    __shared__ float red[128];
    const int tid = threadIdx.x;
    red[tid] = loss_b[tid];
    __syncthreads();
    for (int s = 64; s > 0; s >>= 1) {
        if (tid < s) red[tid] += red[tid + s];
        __syncthreads();
    }
    if (tid == 0) out[0] = red[0] * (1.0f / Bn);
}

extern "C" void kernel_launch(void* const* d_in, const int* in_sizes, int n_in,
                              void* d_out, int out_size, void* d_ws, size_t ws_size,
                              hipStream_t stream) {
    const float* features = (const float*)d_in[0];
    const int*   labels   = (const int*)d_in[1];
    const float* head_w   = (const float*)d_in[2];
    const float* head_b   = (const float*)d_in[3];
    const float* start_t  = (const float*)d_in[4];
    const float* end_t    = (const float*)d_in[5];
    const float* trans    = (const float*)d_in[6];

    float* logits = (float*)d_ws;                        // B*T*L = 12.6 MB
    float* loss_b = logits + (size_t)Bn * Tn * Ln;       // B floats
    float* out    = (float*)d_out;

    crf_logits_wmma<<<(Bn * Tn) / 128, 256, 0, stream>>>(features, head_w, head_b, logits);
    crf_forward<<<Bn, 192, 0, stream>>>(logits, labels, start_t, end_t, trans, loss_b);
    crf_reduce<<<1, 128, 0, stream>>>(loss_b, out);
}